// ChebyConvolution_10660108828938
// MI455X (gfx1250) — compile-verified
//
#include <hip/hip_runtime.h>
#include <hip/hip_bf16.h>

#define N_NODES 50000
#define N_EDGES 1600000
#define FDIM    64
#define K_CHEB  4

typedef float v2f __attribute__((ext_vector_type(2)));
typedef float v8f __attribute__((ext_vector_type(8)));

// ---------------------------------------------------------------------------
// dst[i] = scale * src[i]   (scale=0 -> zero init; scale=-1 -> dst = -src,
// used to fold the "- T_{k-2}" term of the Chebyshev recurrence into the
// accumulator before the scatter-add pass)
// ---------------------------------------------------------------------------
__global__ void cheby_init_kernel(float* __restrict__ dst,
                                  const float* __restrict__ src,
                                  float scale, int n) {
    int i = blockIdx.x * blockDim.x + threadIdx.x;
    if (i < n) dst[i] = scale * src[i];
}

// ---------------------------------------------------------------------------
// SpMM scatter pass: dst[row[e], :] += scale * val[e] * src[col[e], :]
// 64 threads per edge (one feature each) -> fully coalesced 256B gather of
// src row + contiguous f32 atomics into dst row. Feature matrices are
// L2-resident (12.8MB each vs 192MB L2), so this is L2-atomic bound.
// ---------------------------------------------------------------------------
__global__ void cheby_spmm_kernel(const int*   __restrict__ row,
                                  const int*   __restrict__ col,
                                  const float* __restrict__ val,
                                  const float* __restrict__ src,
                                  float*       __restrict__ dst,
                                  float scale) {
    long long t = (long long)blockIdx.x * blockDim.x + threadIdx.x;
    int e = (int)(t >> 6);      // edge index
    int f = (int)(t & 63);      // feature index
    if (e >= N_EDGES) return;
    int r = row[e];
    int c = col[e];
    float v = scale * val[e];
    atomicAdd(&dst[r * FDIM + f], v * src[c * FDIM + f]);
}

// ---------------------------------------------------------------------------
// Dense contraction: out[N,64] = sum_k T_k[N,64] @ W[k,64,64] + bias
// One wave32 per 16x16 output tile using V_WMMA_F32_16X16X4_F32.
//   A tile (16x4 f32): lane l%16 -> row M=l%16; lane/16 selects K-pair {0,1}/{2,3}
//   B tile (4x16 f32): lane l%16 -> col N=l%16; lane/16 selects K-pair
//   C/D (16x16 f32)  : VGPR j, lane half selects M = j or j+8, N = lane%16
// ---------------------------------------------------------------------------
__global__ void cheby_gemm_wmma_kernel(const float* __restrict__ t0,
                                       const float* __restrict__ t1,
                                       const float* __restrict__ t2,
                                       const float* __restrict__ t3,
                                       const float* __restrict__ w,    // [K,64,64]
                                       const float* __restrict__ bias, // [64]
                                       float*       __restrict__ out) {
    const int lane  = threadIdx.x & 31;
    const int wave  = (blockIdx.x * blockDim.x + threadIdx.x) >> 5;
    const int tilesN = FDIM / 16;               // 4 tiles across F_OUT
    const int m0 = (wave / tilesN) * 16;        // node-tile origin
    const int n0 = (wave % tilesN) * 16;        // output-feature-tile origin
    if (m0 >= N_NODES) return;                  // wave-uniform: EXEC stays full

    const int half = lane >> 4;                 // 0: K={0,1}/M=j ; 1: K={2,3}/M=j+8
    const int l15  = lane & 15;

    const float* T[K_CHEB] = { t0, t1, t2, t3 };

    v8f acc = {};
    #pragma unroll
    for (int k = 0; k < K_CHEB; ++k) {
        const float* __restrict__ tk = T[k];
        const float* __restrict__ wk = w + k * FDIM * FDIM;
        #pragma unroll
        for (int kk = 0; kk < FDIM; kk += 4) {
            // A: T_k[m0 + l15, kk + 2*half .. +1]  (contiguous 8B load)
            v2f a = *(const v2f*)(tk + (m0 + l15) * FDIM + kk + 2 * half);
            // B: W_k[kk + 2*half .. +1, n0 + l15]  (two strided loads)
            v2f b;
            b.x = wk[(kk + 2 * half    ) * FDIM + n0 + l15];
            b.y = wk[(kk + 2 * half + 1) * FDIM + n0 + l15];
            acc = __builtin_amdgcn_wmma_f32_16x16x4_f32(
                /*neg_a=*/false, a, /*neg_b=*/false, b,
                /*c_mod=*/(short)0, acc, /*reuse_a=*/false, /*reuse_b=*/false);
        }
    }

    const float bb = bias[n0 + l15];
    #pragma unroll
    for (int j = 0; j < 8; ++j) {
        int m = m0 + j + 8 * half;
        out[m * FDIM + n0 + l15] = acc[j] + bb;
    }
}

// ---------------------------------------------------------------------------
// Launch: T1 = A x ; T2 = 2 A T1 - x ; T3 = 2 A T2 - T1 ; out = sum T_k W_k + b
// ---------------------------------------------------------------------------
extern "C" void kernel_launch(void* const* d_in, const int* in_sizes, int n_in,
                              void* d_out, int out_size, void* d_ws, size_t ws_size,
                              hipStream_t stream) {
    const float* x    = (const float*)d_in[0];
    const int*   arow = (const int*)  d_in[1];
    const int*   acol = (const int*)  d_in[2];
    const float* aval = (const float*)d_in[3];
    const float* wts  = (const float*)d_in[4];
    const float* bias = (const float*)d_in[5];
    float*       out  = (float*)d_out;

    const int NF = N_NODES * FDIM;              // 3.2M floats per term
    float* T1 = (float*)d_ws;
    float* T2 = T1 + NF;
    float* T3 = T2 + NF;

    dim3 blk(256);
    dim3 grid_init((NF + 255) / 256);
    dim3 grid_spmm((unsigned)(((long long)N_EDGES * 64 + 255) / 256));
    const int tiles = (N_NODES / 16) * (FDIM / 16);          // 12500 waves
    dim3 grid_gemm((tiles * 32 + 255) / 256);

    // T1 = A x
    cheby_init_kernel<<<grid_init, blk, 0, stream>>>(T1, x, 0.0f, NF);
    cheby_spmm_kernel<<<grid_spmm, blk, 0, stream>>>(arow, acol, aval, x,  T1, 1.0f);
    // T2 = 2 A T1 - x
    cheby_init_kernel<<<grid_init, blk, 0, stream>>>(T2, x, -1.0f, NF);
    cheby_spmm_kernel<<<grid_spmm, blk, 0, stream>>>(arow, acol, aval, T1, T2, 2.0f);
    // T3 = 2 A T2 - T1
    cheby_init_kernel<<<grid_init, blk, 0, stream>>>(T3, T1, -1.0f, NF);
    cheby_spmm_kernel<<<grid_spmm, blk, 0, stream>>>(arow, acol, aval, T2, T3, 2.0f);
    // out = sum_k T_k @ W_k + bias
    cheby_gemm_wmma_kernel<<<grid_gemm, blk, 0, stream>>>(x, T1, T2, T3, wts, bias, out);
}